// KNNGrouper_13709535609353
// MI455X (gfx1250) — compile-verified
//
#include <hip/hip_runtime.h>
#include <hip/hip_bf16.h>

// Problem constants (match reference)
#define BATCH 8
#define NPTS  32768
#define NGRP  1024
#define KNN   32

typedef __attribute__((ext_vector_type(2))) float v2f;
typedef __attribute__((ext_vector_type(8))) float v8f;

#define FLT_BIG 3.402823466e38f

// ---------------------------------------------------------------------------
// Kernel 1: Farthest Point Sampling. One 1024-thread block per batch.
// min_d lives in 32 VGPRs per thread; argmax via wave32 shuffle butterflies +
// one cross-wave LDS round (2 barriers per FPS round).
// ---------------------------------------------------------------------------
__global__ __launch_bounds__(1024) void fps_kernel(const float* __restrict__ xyz,
                                                   int* __restrict__ fps_idx) {
  const int b = blockIdx.x;
  const int t = threadIdx.x;
  const int lane = t & 31;
  const int wave = t >> 5;
  const float* xb = xyz + (size_t)b * NPTS * 3;

  __shared__ float s_v[32];
  __shared__ int   s_i[32];
  __shared__ int   s_cur;

  float md[32];
#pragma unroll
  for (int j = 0; j < 32; ++j) md[j] = FLT_BIG;

  int cur = 0;
  for (int g = 0; g < NGRP; ++g) {
    if (t == 0) fps_idx[b * NGRP + g] = cur;
    const float px = xb[cur * 3 + 0];
    const float py = xb[cur * 3 + 1];
    const float pz = xb[cur * 3 + 2];

    float bv = -1.0f;
    int   bi = 0x7fffffff;
#pragma unroll
    for (int j = 0; j < 32; ++j) {
      const int p = t + j * 1024;
      const float dx = xb[p * 3 + 0] - px;
      const float dy = xb[p * 3 + 1] - py;
      const float dz = xb[p * 3 + 2] - pz;
      const float d = dx * dx + dy * dy + dz * dz;
      md[j] = fminf(md[j], d);
      if (md[j] > bv) { bv = md[j]; bi = p; }   // j ascending => first index kept on ties
    }
    // wave32 argmax butterfly (lowest index wins ties)
#pragma unroll
    for (int m = 16; m >= 1; m >>= 1) {
      const float ov = __shfl_xor(bv, m, 32);
      const int   oi = __shfl_xor(bi, m, 32);
      if (ov > bv || (ov == bv && oi < bi)) { bv = ov; bi = oi; }
    }
    if (lane == 0) { s_v[wave] = bv; s_i[wave] = bi; }
    __syncthreads();
    if (wave == 0) {
      bv = s_v[lane]; bi = s_i[lane];
#pragma unroll
      for (int m = 16; m >= 1; m >>= 1) {
        const float ov = __shfl_xor(bv, m, 32);
        const int   oi = __shfl_xor(bi, m, 32);
        if (ov > bv || (ov == bv && oi < bi)) { bv = ov; bi = oi; }
      }
      if (lane == 0) s_cur = bi;
    }
    __syncthreads();
    cur = s_cur;
  }
}

// ---------------------------------------------------------------------------
// Branchless sorted insert into a length-32 register-resident list, keyed by
// (d, idx) lexicographic (matches jax.lax.top_k tie-breaking). Caller must
// guarantee (d,p) < (bd[31],bi[31]).  Must stay fully unrolled so bd/bi stay
// in VGPRs (runtime indexing would spill to scratch).
// ---------------------------------------------------------------------------
__device__ __forceinline__ void insert32(float d, int p, float* bd, int* bi) {
#pragma unroll
  for (int j = 31; j >= 1; --j) {
    const bool sh = (d < bd[j - 1]) || (d == bd[j - 1] && p < bi[j - 1]);
    const bool pl = !sh && ((d < bd[j]) || (d == bd[j] && p < bi[j]));
    const float nd = sh ? bd[j - 1] : (pl ? d : bd[j]);
    const int   ni = sh ? bi[j - 1] : (pl ? p : bi[j]);
    bd[j] = nd; bi[j] = ni;
  }
  if ((d < bd[0]) || (d == bd[0] && p < bi[0])) { bd[0] = d; bi[0] = p; }
}

// ---------------------------------------------------------------------------
// Kernel 2: WMMA distance GEMM + fused top-32 selection + gather.
// Block = 256 threads (8 waves); each block owns a 16-center tile.
// Wave w scans points [w*4096, (w+1)*4096) in 32-point steps:
//   two V_WMMA_F32_16X16X4_F32 ops per step (A = [-2q, q^2] aug, B = [k, 1]).
// Lanes 0-15 select over even 16-point subtiles, lanes 16-31 over odd ones.
// ALL merges go through LDS with runtime loops so the ~300-instruction insert
// body exists exactly once per loop (fits the hot path in the WGP I-cache).
// ---------------------------------------------------------------------------
__global__ __launch_bounds__(256) void knn_kernel(const float* __restrict__ xyz,
                                                  const int* __restrict__ fps,
                                                  float* __restrict__ out) {
  const int b    = blockIdx.x >> 6;           // 64 tiles per batch (1024/16)
  const int tile = blockIdx.x & 63;
  const int lane = threadIdx.x & 31;
  const int w    = threadIdx.x >> 5;
  const float* xb = xyz + (size_t)b * NPTS * 3;

  __shared__ float s_tile[8][2][16][16];      // per-wave 16x16 C tiles (16 KB)
  __shared__ float s_cd[16][8][32];           // per-center per-wave top-32 dists (16 KB)
  __shared__ int   s_ci[16][8][32];           // indices (16 KB)

  // Build augmented A (16 centers x 4): row = [-2cx, -2cy, -2cz, |c|^2]
  // A VGPR layout: v0 = K0 (lanes 0-15) / K2 (lanes 16-31); v1 = K1 / K3.
  const int cm   = lane & 15;
  const int cidx = fps[b * NGRP + tile * 16 + cm];
  const float cx = xb[cidx * 3 + 0];
  const float cy = xb[cidx * 3 + 1];
  const float cz = xb[cidx * 3 + 2];
  const float q2 = cx * cx + cy * cy + cz * cz;
  v2f a;
  a.x = (lane < 16) ? (-2.0f * cx) : (-2.0f * cz);
  a.y = (lane < 16) ? (-2.0f * cy) : q2;

  float bd[32]; int bi[32];
#pragma unroll
  for (int j = 0; j < 32; ++j) { bd[j] = FLT_BIG; bi[j] = 0x7fffffff; }

  const int wbase = w * 4096;
  for (int it = 0; it < 128; ++it) {
    const int pbase = wbase + it * 32;
    __builtin_prefetch(xb + (size_t)(pbase + 96) * 3, 0, 3);  // near-cache prefetch

#pragma unroll
    for (int tsub = 0; tsub < 2; ++tsub) {
      // B (4 x 16): col = [kx, ky, kz, 1]; v0 = K0 / K1, v1 = K2 / K3
      const int pn = pbase + tsub * 16 + (lane & 15);
      const float kx = xb[pn * 3 + 0];
      const float ky = xb[pn * 3 + 1];
      const float kz = xb[pn * 3 + 2];
      const float k2 = kx * kx + ky * ky + kz * kz;
      v2f bb;
      bb.x = (lane < 16) ? kx : ky;
      bb.y = (lane < 16) ? kz : 1.0f;

      v8f c = {};
      c = __builtin_amdgcn_wmma_f32_16x16x4_f32(false, a, false, bb,
                                                (short)0, c, false, false);
      // C layout: VGPR r -> (M=r, N=lane) lanes 0-15 ; (M=r+8, N=lane-16)
      const int n = lane & 15;
#pragma unroll
      for (int r = 0; r < 8; ++r) {
        const int m = (lane < 16) ? r : (r + 8);
        s_tile[w][tsub][m][n] = c[r] + k2;   // full d^2 = q^2 - 2qk + k^2
      }
    }
    asm volatile("s_wait_dscnt 0" ::: "memory");   // intra-wave LDS visibility

    // lanes 0-15: subtile 0 of center=lane; lanes 16-31: subtile 1 of center=lane-16
    const int selm = lane & 15;
    const int selt = lane >> 4;
    const int psel = pbase + selt * 16;
    const float* row = &s_tile[w][selt][selm][0];
#pragma unroll 1
    for (int cc = 0; cc < 16; ++cc) {
      const float d = row[cc];
      const int   p = psel + cc;
      if (d < bd[31] || (d == bd[31] && p < bi[31])) insert32(d, p, bd, bi);
    }
  }

  // Lane-pair merge via LDS (no shuffle-unrolled insert clones):
  // 1) lanes 16-31 park their lists in s_cd/s_ci (cheap unrolled stores)
  if (lane >= 16) {
#pragma unroll
    for (int j = 0; j < 32; ++j) { s_cd[lane - 16][w][j] = bd[j]; s_ci[lane - 16][w][j] = bi[j]; }
  }
  asm volatile("s_wait_dscnt 0" ::: "memory");
  // 2) lanes 0-15 merge them in a runtime loop (single insert body)
  if (lane < 16) {
#pragma unroll 1
    for (int j = 0; j < 32; ++j) {
      const float od = s_cd[lane][w][j];
      const int   oi = s_ci[lane][w][j];
      if (od < bd[31] || (od == bd[31] && oi < bi[31])) insert32(od, oi, bd, bi);
    }
  }
  asm volatile("s_wait_dscnt 0" ::: "memory");
  // 3) lanes 0-15 publish the wave-merged list
  if (lane < 16) {
#pragma unroll
    for (int j = 0; j < 32; ++j) { s_cd[lane][w][j] = bd[j]; s_ci[lane][w][j] = bi[j]; }
  }
  __syncthreads();

  // Final cross-wave merge + output (wave 0, lanes 0-15, one center each).
  // Runtime loops: candidates come from LDS, insert body exists once.
  if (w == 0 && lane < 16) {
    float fd[32]; int fi[32];
#pragma unroll
    for (int j = 0; j < 32; ++j) { fd[j] = s_cd[lane][0][j]; fi[j] = s_ci[lane][0][j]; }
#pragma unroll 1
    for (int ww = 1; ww < 8; ++ww) {
#pragma unroll 1
      for (int j = 0; j < 32; ++j) {
        const float d = s_cd[lane][ww][j];
        const int   p = s_ci[lane][ww][j];
        if (d < fd[31] || (d == fd[31] && p < fi[31])) insert32(d, p, fd, fi);
      }
    }
    const int   g     = tile * 16 + lane;
    const size_t base = ((size_t)b * NGRP + g) * KNN;
    float* out_idx = out;                                   // indices as float (exact < 2^24)
    float* out_nbr = out + (size_t)BATCH * NGRP * KNN;
#pragma unroll
    for (int k = 0; k < KNN; ++k) {
      const int p = fi[k];
      out_idx[base + k] = (float)(p + b * NPTS);
      const float* sp = xb + (size_t)p * 3;
      float* dp = out_nbr + (base + k) * 3;
      dp[0] = sp[0]; dp[1] = sp[1]; dp[2] = sp[2];
    }
  }
}

extern "C" void kernel_launch(void* const* d_in, const int* in_sizes, int n_in,
                              void* d_out, int out_size, void* d_ws, size_t ws_size,
                              hipStream_t stream) {
  (void)in_sizes; (void)n_in; (void)out_size; (void)ws_size;
  const float* xyz = (const float*)d_in[0];
  float* out = (float*)d_out;
  int* fps = (int*)d_ws;                       // B*G ints = 32 KB of scratch

  fps_kernel<<<dim3(BATCH), dim3(1024), 0, stream>>>(xyz, fps);
  knn_kernel<<<dim3(BATCH * (NGRP / 16)), dim3(256), 0, stream>>>(xyz, fps, out);
}